// GaussianRenderer_429496729774
// MI455X (gfx1250) — compile-verified
//
#include <hip/hip_runtime.h>
#include <hip/hip_bf16.h>
#include <math.h>

// ---------------------------------------------------------------------------
// Gaussian point-splat renderer for MI455X (gfx1250, wave32).
//   pass 0: init z-buffer (+inf bits) and index buffer (N)
//   pass 1: WMMA point transform -> project -> atomicMin(z-bits)   [16 pts/wave]
//   pass 2: WMMA point transform -> project -> atomicMin(idx) for depth winners
//   pass 3: per-pixel resolve: gather sigmoid(colour[:3]) -> (3,H,W) canvas
// The 4x4 view transform is done with V_WMMA_F32_16X16X4_F32:
//   A(16x4) = top-3 rows of E (R | t), rows 3..15 zero
//   B(4x16) = 16 homogeneous points as columns (x,y,z,1)
//   D(16x16): lane j (j<16) holds D[0][j],D[1][j],D[2][j] = x',y',z' of point j
// ---------------------------------------------------------------------------

typedef __attribute__((ext_vector_type(2))) float v2f;
typedef __attribute__((ext_vector_type(8))) float v8f;

#define H_IMG 352
#define W_IMG 640
#define NPIX  (H_IMG * W_IMG)

// --- WMMA transform of 16 points (base..base+15) -------------------------------
__device__ __forceinline__ v8f gr_transform16(const float* __restrict__ means,
                                              const float* __restrict__ E,
                                              long long base, int lane,
                                              long long N)
{
    const int m = lane & 15;
    long long p = base + m;
    if (p >= N) p = N - 1;              // clamp (N%16==0 in practice)

    // B matrix 4x16 (32-bit): VGPR0 = rows {0,2}, VGPR1 = rows {1,3};
    // lanes 0-15 -> rows 0/1 (x,y), lanes 16-31 -> rows 2/3 (z,1).
    v2f b;
    if (lane < 16) {
        b[0] = means[p * 3 + 0];
        b[1] = means[p * 3 + 1];
    } else {
        b[0] = means[p * 3 + 2];
        b[1] = 1.0f;
    }

    // A matrix 16x4 (32-bit): lanes 0-15 -> K=0/1, lanes 16-31 -> K=2/3.
    // Row i (<3) = [E[i][0], E[i][1], E[i][2], E[i][3]]; rows 3..15 zero.
    v2f a;
    a[0] = 0.0f;
    a[1] = 0.0f;
    if (m < 3) {
        a[0] = (lane < 16) ? E[m * 4 + 0] : E[m * 4 + 2];
        a[1] = (lane < 16) ? E[m * 4 + 1] : E[m * 4 + 3];
    }

    v8f c = {};
    // D = A x B + 0   (emits v_wmma_f32_16x16x4_f32)
    return __builtin_amdgcn_wmma_f32_16x16x4_f32(
        /*neg_a=*/false, a, /*neg_b=*/false, b,
        /*c_mod=*/(short)0, c, /*reuse_a=*/false, /*reuse_b=*/false);
}

// --- projection matching the reference semantics -------------------------------
__device__ __forceinline__ bool gr_project(float x, float y, float z,
                                           float fx, float fy, float cx, float cy,
                                           int& pix)
{
    const bool  valid = z > 0.1f;
    const float zs    = valid ? z : 1.0f;
    const int u = (int)(x / zs * fx + cx);   // trunc-toward-zero == astype(int32)
    const int v = (int)(y / zs * fy + cy);
    pix = v * W_IMG + u;
    return valid && (u >= 0) && (u < W_IMG) && (v >= 0) && (v < H_IMG);
}

// --- pass 0: buffer init -------------------------------------------------------
__global__ void gr_init(int* __restrict__ zbits, int* __restrict__ idxbuf, int N)
{
    const int p = blockIdx.x * blockDim.x + threadIdx.x;
    if (p < NPIX) {
        zbits[p]  = 0x7F800000;   // +inf
        idxbuf[p] = N;
    }
}

// --- pass 1: z-buffer scatter --------------------------------------------------
__global__ void gr_zpass(const float* __restrict__ means,
                         const float* __restrict__ Kc,
                         const float* __restrict__ E,
                         int* __restrict__ zbits, long long N)
{
    const int lane = threadIdx.x & 31;
    const long long wave = ((long long)blockIdx.x * blockDim.x + threadIdx.x) >> 5;
    const long long base = wave * 16;
    if (base >= N) return;                   // wave-uniform: EXEC all-ones at WMMA

    const v8f d = gr_transform16(means, E, base, lane, N);

    if (lane < 16 && (base + lane) < N) {
        const float x = d[0], y = d[1], z = d[2];
        int pix;
        if (gr_project(x, y, z, Kc[0], Kc[4], Kc[2], Kc[5], pix)) {
            // positive floats compare like their int bit patterns
            atomicMin(zbits + pix, __float_as_int(z));
        }
    }
}

// --- pass 2: winner-index scatter ---------------------------------------------
__global__ void gr_ipass(const float* __restrict__ means,
                         const float* __restrict__ Kc,
                         const float* __restrict__ E,
                         const float* __restrict__ zbuf,
                         int* __restrict__ idxbuf, long long N)
{
    const int lane = threadIdx.x & 31;
    const long long wave = ((long long)blockIdx.x * blockDim.x + threadIdx.x) >> 5;
    const long long base = wave * 16;
    if (base >= N) return;

    const v8f d = gr_transform16(means, E, base, lane, N);

    if (lane < 16 && (base + lane) < N) {
        const float x = d[0], y = d[1], z = d[2];
        int pix;
        if (gr_project(x, y, z, Kc[0], Kc[4], Kc[2], Kc[5], pix)) {
            if (z <= zbuf[pix]) {
                atomicMin(idxbuf + pix, (int)(base + lane));
            }
        }
    }
}

// --- pass 3: per-pixel resolve -------------------------------------------------
__global__ void gr_resolve(const int* __restrict__ idxbuf,
                           const float* __restrict__ col,
                           float* __restrict__ out, int N)
{
    const int p = blockIdx.x * blockDim.x + threadIdx.x;
    if (p >= NPIX) return;

    const int idx = idxbuf[p];
    float r = 0.0f, g = 0.0f, b = 0.0f;
    if (idx < N) {
        const float* c = col + (long long)idx * 12;  // SH_C = 12, use first 3
        r = 1.0f / (1.0f + expf(-c[0]));
        g = 1.0f / (1.0f + expf(-c[1]));
        b = 1.0f / (1.0f + expf(-c[2]));
    }
    // canvas.T.reshape(3,H,W): channel-major
    out[p]            = r;
    out[NPIX + p]     = g;
    out[2 * NPIX + p] = b;
}

extern "C" void kernel_launch(void* const* d_in, const int* in_sizes, int n_in,
                              void* d_out, int out_size, void* d_ws, size_t ws_size,
                              hipStream_t stream)
{
    // setup_inputs() order: means3D, colours, K, E_v2c, H, W, item
    const float* means = (const float*)d_in[0];
    const float* col   = (const float*)d_in[1];
    const float* Kc    = (const float*)d_in[2];
    const float* E     = (const float*)d_in[3];
    const long long N  = (long long)in_sizes[0] / 3;

    int* zbits  = (int*)d_ws;          // NPIX ints (float bits of z-buffer)
    int* idxbuf = zbits + NPIX;        // NPIX ints
    float* out  = (float*)d_out;       // 3 * NPIX floats

    const dim3 blk(256);
    const int pixBlocks = (NPIX + 255) / 256;

    gr_init<<<pixBlocks, blk, 0, stream>>>(zbits, idxbuf, (int)N);

    // 16 points per wave, 8 waves per 256-thread block -> 128 points per block
    const long long waves   = (N + 15) / 16;
    const long long ptBlock = (waves * 32 + 255) / 256;

    gr_zpass<<<(int)ptBlock, blk, 0, stream>>>(means, Kc, E, zbits, N);
    gr_ipass<<<(int)ptBlock, blk, 0, stream>>>(means, Kc, E,
                                               (const float*)zbits, idxbuf, N);
    gr_resolve<<<pixBlocks, blk, 0, stream>>>(idxbuf, col, out, (int)N);
}